// HierarchicalSS2D_33389075759204
// MI455X (gfx1250) — compile-verified
//
#include <hip/hip_runtime.h>
#include <math.h>

// ---------------------------------------------------------------------------
// HierarchicalSS2D for MI455X (gfx1250, wave32).
// - GEMMs: V_WMMA_F32_16X16X4_F32 (full fp32, matches f32 reference).
// - Selective scan: two-phase kernel. Phase 1 (parallel) computes
//   Delta=softplus(...) and gathers u into a 148 KB LDS tile (CDNA5 WGP has
//   320 KB LDS). Phase 2 (serial over L=2304) is reduced to ~one FMA on the
//   recurrence chain; exp() and the lane reduction overlap across unrolled
//   iterations. global_prefetch keeps streaming B/C lines ahead.
// ---------------------------------------------------------------------------

#define DM   96
#define DI   192
#define NS   16
#define DR   6
#define BB   2
#define HH   48
#define WWD  48
#define LL   (HH * WWD)     // 2304
#define MTOT (BB * LL)      // 4608

#define SCAN_CH       8                    // channels per scan block
#define SCAN_THREADS  (SCAN_CH * 16)       // 128 threads = 4 waves
#define SCAN_STRIDE   (LL + 8)             // pad to spread LDS banks
#define SCAN_LDS_BYTES (2u * SCAN_CH * SCAN_STRIDE * sizeof(float))  // ~148 KB

typedef float v2f __attribute__((ext_vector_type(2)));
typedef float v8f __attribute__((ext_vector_type(8)));

__device__ __forceinline__ v8f wmma_f32_16x16x4(v2f a, v2f b, v8f c) {
  // (neg_a, A, neg_b, B, c_mod, C, reuse_a, reuse_b)
  return __builtin_amdgcn_wmma_f32_16x16x4_f32(false, a, false, b,
                                               (short)0, c, false, false);
}

// Direction permutation (involution): k0=row-major, k1=transpose,
// k2=reverse, k3=transpose(reverse).
__device__ __forceinline__ int sigma_map(int k, int l) {
  if (k == 0) return l;
  if (k == 1) return (l % WWD) * WWD + (l / WWD);
  if (k == 2) return (LL - 1) - l;
  int lf = (LL - 1) - l;
  return (lf % WWD) * WWD + (lf / WWD);
}

__device__ __forceinline__ float silu_f(float x) {
  return x / (1.0f + __expf(-x));
}

// ---------------------------------------------------------------------------
// K1: input projection  xz[b,l,:] = x[b,l,:] @ in_w^T   (M=4608, N=384, K=96)
// o<192 -> xi_pre[br,b,c,l]  (channel-major for conv/scan),  o>=192 -> z[br,b,l,c]
// ---------------------------------------------------------------------------
__global__ void k_inproj(const float* __restrict__ xs, const float* __restrict__ xt,
                         const float* __restrict__ w_sem, const float* __restrict__ w_str,
                         float* __restrict__ xi_pre, float* __restrict__ zbuf) {
  const int lane = threadIdx.x & 31;
  const int half = lane >> 4;
  const int col  = lane & 15;
  const int m0 = blockIdx.x * 16;
  const int n0 = blockIdx.y * 16;
  const int br = blockIdx.z;
  const float* __restrict__ X = br ? xt : xs;          // (MTOT, DM)
  const float* __restrict__ W = br ? w_str : w_sem;    // (2*DI, DM)

  v8f acc = {};
  for (int kb = 0; kb < DM; kb += 4) {
    const int kk = kb + 2 * half;
    v2f a, b;
    a.x = X[(size_t)(m0 + col) * DM + kk];
    a.y = X[(size_t)(m0 + col) * DM + kk + 1];
    b.x = W[(size_t)(n0 + col) * DM + kk];
    b.y = W[(size_t)(n0 + col) * DM + kk + 1];
    acc = wmma_f32_16x16x4(a, b, acc);
  }
#pragma unroll
  for (int i = 0; i < 8; ++i) {
    const int s = m0 + i + 8 * half;
    const int o = n0 + col;
    const float v = acc[i];
    const int b = s / LL, l = s % LL;
    if (o < DI)
      xi_pre[(((size_t)br * BB + b) * DI + o) * LL + l] = v;
    else
      zbuf[(((size_t)br * BB + b) * LL + l) * DI + (o - DI)] = v;
  }
}

// ---------------------------------------------------------------------------
// K2: depthwise 3x3 conv (SAME, zero pad) + bias + SiLU
// ---------------------------------------------------------------------------
__global__ void k_dwconv(const float* __restrict__ xi_pre,
                         const float* __restrict__ cw_sem, const float* __restrict__ cb_sem,
                         const float* __restrict__ cw_str, const float* __restrict__ cb_str,
                         float* __restrict__ xc) {
  const int idx = blockIdx.x * blockDim.x + threadIdx.x;
  const int total = 2 * BB * DI * LL;
  if (idx >= total) return;
  int t = idx;
  const int wp = t % WWD; t /= WWD;
  const int hp = t % HH;  t /= HH;
  const int c  = t % DI;  t /= DI;
  const int b  = t % BB;
  const int br = t / BB;
  const float* __restrict__ w9 = (br ? cw_str : cw_sem) + (size_t)c * 9;
  const float bias = (br ? cb_str : cb_sem)[c];
  const float* __restrict__ src = xi_pre + (((size_t)br * BB + b) * DI + c) * LL;
  float acc = bias;
#pragma unroll
  for (int di = 0; di < 3; ++di) {
    const int h = hp + di - 1;
    if ((unsigned)h >= (unsigned)HH) continue;
#pragma unroll
    for (int dj = 0; dj < 3; ++dj) {
      const int w = wp + dj - 1;
      if ((unsigned)w >= (unsigned)WWD) continue;
      acc += w9[di * 3 + dj] * src[h * WWD + w];
    }
  }
  xc[idx] = silu_f(acc);
}

// ---------------------------------------------------------------------------
// K3: directional projection  x_dbl[b,k,r,l] = sum_d xproj_w[k,r,d]*xc[b,d,sigma_k(l)]
// WMMA GEMM, M over L (2304), N over 38 rows (3 padded 16-tiles), K=192.
// Rows 0..5 -> dtr (l-major, pad 8); 6..21 -> Bs[l,16]; 22..37 -> Cs[l,16].
// ---------------------------------------------------------------------------
__global__ void k_xproj(const float* __restrict__ xc,
                        const float* __restrict__ xp_sem, const float* __restrict__ xp_str,
                        float* __restrict__ dtr, float* __restrict__ Bsb,
                        float* __restrict__ Csb) {
  const int lane = threadIdx.x & 31;
  const int half = lane >> 4;
  const int col  = lane & 15;
  const int m0 = blockIdx.x * 16;
  const int n0 = blockIdx.y * 16;
  const int zz = blockIdx.z;          // 0..15
  const int br = zz >> 3;
  const int b  = (zz >> 2) & 1;
  const int k  = zz & 3;
  const float* __restrict__ W = (br ? xp_str : xp_sem);               // (4,38,DI)
  const float* __restrict__ U = xc + ((size_t)br * BB + b) * DI * LL; // (DI, LL)

  const int m_src = sigma_map(k, m0 + col);   // gathered A row
  const int r = n0 + col;
  const bool rok = (r < DR + 2 * NS);

  v8f acc = {};
  for (int kb = 0; kb < DI; kb += 4) {
    const int kk = kb + 2 * half;
    v2f a, bf;
    a.x = U[(size_t)(kk)     * LL + m_src];
    a.y = U[(size_t)(kk + 1) * LL + m_src];
    bf.x = rok ? W[((size_t)k * (DR + 2 * NS) + r) * DI + kk]     : 0.0f;
    bf.y = rok ? W[((size_t)k * (DR + 2 * NS) + r) * DI + kk + 1] : 0.0f;
    acc = wmma_f32_16x16x4(a, bf, acc);
  }
  const size_t p = ((size_t)br * BB + b) * 4 + k;
#pragma unroll
  for (int i = 0; i < 8; ++i) {
    const int l = m0 + i + 8 * half;
    const float v = acc[i];
    if (r < DR)
      dtr[(p * LL + l) * 8 + r] = v;
    else if (r < DR + NS)
      Bsb[(p * LL + l) * 16 + (r - DR)] = v;
    else if (r < DR + 2 * NS)
      Csb[(p * LL + l) * 16 + (r - DR - NS)] = v;
  }
}

// ---------------------------------------------------------------------------
// K4: selective scan, two-phase.
// Block = 128 threads (4 waves) owns SCAN_CH=8 channels of one (b,k).
// Phase 1 (parallel): Delta = softplus(dt_w . dtr + dt_b) and u = xc[c, sigma_k(l)]
//   for all (c,l) of the block -> LDS (148 KB of the WGP's 320 KB).
// Phase 2 (serial over l): lane n of each 16-lane half holds state x[n];
//   x = exp(Delta*A[n])*x + (Delta*B[n,l]*u)  -- only the FMA is on the chain.
//   y(sigma_k(l)) = cross-lane sum(x*C) + u*D.  Direction k owns its y4 slice
//   (bijective scatter, no atomics -> deterministic).
// ---------------------------------------------------------------------------
__global__ void k_scan(const float* __restrict__ xc_br,
                       const float* __restrict__ dtr, const float* __restrict__ Bsb,
                       const float* __restrict__ Csb,
                       const float* __restrict__ dt_w, const float* __restrict__ dt_b,
                       const float* __restrict__ A_logs, const float* __restrict__ Ds,
                       const float* __restrict__ x0, float* __restrict__ last,
                       float* __restrict__ y4, int br) {
  extern __shared__ float smem[];
  float* __restrict__ sdelta = smem;                         // [SCAN_CH][SCAN_STRIDE]
  float* __restrict__ su     = smem + SCAN_CH * SCAN_STRIDE; // [SCAN_CH][SCAN_STRIDE]

  const int tid = threadIdx.x;
  const int cbase = blockIdx.x * SCAN_CH;
  const int k = blockIdx.y;
  const int b = blockIdx.z;
  const size_t p = ((size_t)br * BB + b) * 4 + k;
  const float* __restrict__ dtrp = dtr + p * LL * 8;

  // ---- Phase 1: parallel Delta + gathered-u staging into LDS ----
  for (int cl = 0; cl < SCAN_CH; ++cl) {
    const int c  = cbase + cl;
    const int kc = k * DI + c;
    const float dtb = dt_b[kc];                 // wave-uniform -> scalar regs
    const float w0 = dt_w[(size_t)kc * DR + 0], w1 = dt_w[(size_t)kc * DR + 1];
    const float w2 = dt_w[(size_t)kc * DR + 2], w3 = dt_w[(size_t)kc * DR + 3];
    const float w4 = dt_w[(size_t)kc * DR + 4], w5 = dt_w[(size_t)kc * DR + 5];
    const float* __restrict__ up = xc_br + ((size_t)b * DI + c) * LL;
    for (int l = tid; l < LL; l += SCAN_THREADS) {
      const float* dl = dtrp + (size_t)l * 8;
      float dts = w0 * dl[0] + w1 * dl[1] + w2 * dl[2] +
                  w3 * dl[3] + w4 * dl[4] + w5 * dl[5] + dtb;
      const float delta = (dts > 20.0f) ? dts : log1pf(__expf(dts));
      sdelta[cl * SCAN_STRIDE + l] = delta;
      su[cl * SCAN_STRIDE + l]     = up[sigma_map(k, l)];
    }
  }
  __syncthreads();

  // ---- Phase 2: serial recurrence ----
  const int lane = tid & 31;
  const int wid  = tid >> 5;            // 0..3
  const int half = lane >> 4;
  const int n    = lane & 15;
  const int cl   = wid * 2 + half;      // 0..7
  const int c    = cbase + cl;
  const int kc   = k * DI + c;

  const float An = -__expf(A_logs[(size_t)kc * NS + n]);
  const float Dc = Ds[kc];
  float xstate = x0 ? x0[(((size_t)b * 4 + k) * DI + c) * NS + n] : 0.0f;

  const float* __restrict__ Bp = Bsb + p * LL * 16;
  const float* __restrict__ Cp = Csb + p * LL * 16;
  const float* __restrict__ srd = sdelta + (size_t)cl * SCAN_STRIDE;
  const float* __restrict__ sru = su + (size_t)cl * SCAN_STRIDE;
  float* __restrict__ yp = y4 + (((size_t)k * BB + b) * DI + c) * LL;

#pragma unroll 4
  for (int l = 0; l < LL; ++l) {
    // keep streaming B/C lines ahead of the serial consumer
    __builtin_prefetch(Bp + (size_t)(l + 96) * 16, 0, 1);
    __builtin_prefetch(Cp + (size_t)(l + 96) * 16, 0, 1);
    const float delta = srd[l];                 // LDS broadcast
    const float u     = sru[l];
    const float bv = Bp[(size_t)l * 16 + n];    // wave reads one 64B line
    const float cv = Cp[(size_t)l * 16 + n];
    const float aa = __expf(delta * An);        // off the xstate chain
    const float bu = delta * bv * u;            // off the xstate chain
    xstate = fmaf(aa, xstate, bu);              // the only serial dependence
    float prod = xstate * cv;
#pragma unroll
    for (int o = 1; o < 16; o <<= 1)            // xor masks stay inside half
      prod += __shfl_xor(prod, o, 32);
    if (n == 0) yp[sigma_map(k, l)] = prod + u * Dc;
  }
  last[(((size_t)b * 4 + k) * DI + c) * NS + n] = xstate;
}

// ---------------------------------------------------------------------------
// K5: sum 4 directions, channel LayerNorm over 384, gate with SiLU(z) -> g
// ---------------------------------------------------------------------------
__global__ void k_normgate(const float* __restrict__ y4_sem, const float* __restrict__ y4_str,
                           const float* __restrict__ zbuf,
                           const float* __restrict__ gam, const float* __restrict__ bet,
                           float* __restrict__ g) {
  const int bm = blockIdx.x;            // 0..MTOT-1
  const int b = bm / LL, m = bm % LL;
  const int tid = threadIdx.x;          // 128 threads, 3 channels each
  __shared__ float s1[128];
  __shared__ float s2[128];
  float vals[3];
  float lsum = 0.0f, lsq = 0.0f;
#pragma unroll
  for (int j = 0; j < 3; ++j) {
    const int cc = tid + j * 128;
    const float* __restrict__ y4 = (cc < DI) ? y4_sem : y4_str;
    const int c = (cc < DI) ? cc : cc - DI;
    float v = 0.0f;
#pragma unroll
    for (int k = 0; k < 4; ++k)
      v += y4[(((size_t)k * BB + b) * DI + c) * LL + m];
    vals[j] = v; lsum += v; lsq += v * v;
  }
  s1[tid] = lsum; s2[tid] = lsq;
  __syncthreads();
  for (int s = 64; s > 0; s >>= 1) {
    if (tid < s) { s1[tid] += s1[tid + s]; s2[tid] += s2[tid + s]; }
    __syncthreads();
  }
  const float mu   = s1[0] * (1.0f / (2 * DI));
  const float var  = s2[0] * (1.0f / (2 * DI)) - mu * mu;
  const float rstd = rsqrtf(var + 1e-5f);
#pragma unroll
  for (int j = 0; j < 3; ++j) {
    const int cc = tid + j * 128;
    const int zbr = (cc < DI) ? 0 : 1;
    const int zc  = (cc < DI) ? cc : cc - DI;
    const float zz = zbuf[(((size_t)zbr * BB + b) * LL + m) * DI + zc];
    const float v = (vals[j] - mu) * rstd * gam[cc] + bet[cc];
    g[(size_t)bm * (2 * DI) + cc] = v * silu_f(zz);
  }
}

// ---------------------------------------------------------------------------
// K6: output projection  out[s,o] = sum_c g[s,c]*out_w[o,c]  (M=4608,N=192,K=384)
// ---------------------------------------------------------------------------
__global__ void k_outproj(const float* __restrict__ g, const float* __restrict__ ow,
                          float* __restrict__ out) {
  const int lane = threadIdx.x & 31;
  const int half = lane >> 4;
  const int col  = lane & 15;
  const int m0 = blockIdx.x * 16;
  const int n0 = blockIdx.y * 16;
  v8f acc = {};
  for (int kb = 0; kb < 2 * DI; kb += 4) {
    const int kk = kb + 2 * half;
    v2f a, bf;
    a.x = g[(size_t)(m0 + col) * (2 * DI) + kk];
    a.y = g[(size_t)(m0 + col) * (2 * DI) + kk + 1];
    bf.x = ow[(size_t)(n0 + col) * (2 * DI) + kk];
    bf.y = ow[(size_t)(n0 + col) * (2 * DI) + kk + 1];
    acc = wmma_f32_16x16x4(a, bf, acc);
  }
#pragma unroll
  for (int i = 0; i < 8; ++i) {
    const int s = m0 + i + 8 * half;
    out[(size_t)s * (2 * DM) + (n0 + col)] = acc[i];
  }
}

// ---------------------------------------------------------------------------
extern "C" void kernel_launch(void* const* d_in, const int* in_sizes, int n_in,
                              void* d_out, int out_size, void* d_ws, size_t ws_size,
                              hipStream_t stream) {
  (void)in_sizes; (void)n_in; (void)out_size; (void)ws_size;
  const float* sem_x      = (const float*)d_in[0];
  const float* str_x      = (const float*)d_in[1];
  const float* sem_in_w   = (const float*)d_in[2];
  const float* sem_conv_w = (const float*)d_in[3];
  const float* sem_conv_b = (const float*)d_in[4];
  const float* sem_xproj  = (const float*)d_in[5];
  const float* sem_dt_w   = (const float*)d_in[6];
  const float* sem_dt_b   = (const float*)d_in[7];
  const float* sem_A      = (const float*)d_in[8];
  const float* sem_Ds     = (const float*)d_in[9];
  const float* str_in_w   = (const float*)d_in[10];
  const float* str_conv_w = (const float*)d_in[11];
  const float* str_conv_b = (const float*)d_in[12];
  const float* str_xproj  = (const float*)d_in[13];
  const float* str_dt_w   = (const float*)d_in[14];
  const float* str_dt_b   = (const float*)d_in[15];
  const float* str_A      = (const float*)d_in[16];
  const float* str_Ds     = (const float*)d_in[17];
  const float* norm_g     = (const float*)d_in[18];
  const float* norm_b     = (const float*)d_in[19];
  const float* out_w      = (const float*)d_in[20];

  float* ws = (float*)d_ws;
  size_t off = 0;
  float* xi_pre = ws + off; off += (size_t)2 * BB * DI * LL;   // (br,b,C,L)
  float* zbuf   = ws + off; off += (size_t)2 * BB * LL * DI;   // (br,b,L,C)
  float* xc     = ws + off; off += (size_t)2 * BB * DI * LL;   // (br,b,C,L)
  float* dtr    = ws + off; off += (size_t)2 * BB * 4 * LL * 8;
  float* Bsb    = ws + off; off += (size_t)2 * BB * 4 * LL * 16;
  float* Csb    = ws + off; off += (size_t)2 * BB * 4 * LL * 16;
  float* y4_sem = ws + off; off += (size_t)4 * BB * DI * LL;   // (k,b,C,L)
  float* y4_str = ws + off; off += (size_t)4 * BB * DI * LL;
  float* lastb  = ws + off; off += (size_t)BB * 4 * DI * NS;
  float* gbuf   = ws + off; off += (size_t)BB * LL * 2 * DI;   // ~62 MB total

  const dim3 b32(32, 1, 1);

  // 1) input projection (both branches)
  k_inproj<<<dim3(MTOT / 16, (2 * DI) / 16, 2), b32, 0, stream>>>(
      sem_x, str_x, sem_in_w, str_in_w, xi_pre, zbuf);

  // 2) depthwise conv + SiLU
  const int nconv = 2 * BB * DI * LL;
  k_dwconv<<<dim3((nconv + 255) / 256), dim3(256), 0, stream>>>(
      xi_pre, sem_conv_w, sem_conv_b, str_conv_w, str_conv_b, xc);

  // 3) directional projections (both branches, all b,k)
  k_xproj<<<dim3(LL / 16, 3, 16), b32, 0, stream>>>(
      xc, sem_xproj, str_xproj, dtr, Bsb, Csb);

  // 4) scans: semantic first (x0 = 0, writes last), then structural (x0 = last)
  const dim3 gscan(DI / SCAN_CH, 4, BB);
  k_scan<<<gscan, dim3(SCAN_THREADS), SCAN_LDS_BYTES, stream>>>(
      xc, dtr, Bsb, Csb, sem_dt_w, sem_dt_b, sem_A, sem_Ds,
      nullptr, lastb, y4_sem, 0);
  k_scan<<<gscan, dim3(SCAN_THREADS), SCAN_LDS_BYTES, stream>>>(
      xc + (size_t)BB * DI * LL, dtr, Bsb, Csb, str_dt_w, str_dt_b, str_A, str_Ds,
      lastb, lastb, y4_str, 1);

  // 5) direction-sum + LayerNorm + SiLU gate
  k_normgate<<<dim3(MTOT), dim3(128), 0, stream>>>(
      y4_sem, y4_str, zbuf, norm_g, norm_b, gbuf);

  // 6) output projection
  k_outproj<<<dim3(MTOT / 16, (2 * DM) / 16), b32, 0, stream>>>(
      gbuf, out_w, (float*)d_out);
}